// SetGNN_78735340470808
// MI455X (gfx1250) — compile-verified
//
#include <hip/hip_runtime.h>

typedef __attribute__((ext_vector_type(2))) float v2f;
typedef __attribute__((ext_vector_type(8))) float v8f;

#define DFEAT 64
#define LDSTRIDE 68   // 16-row LDS tile stride (floats); 68%64=4 -> spreads banks

// ---------------------------------------------------------------------------
// Fused 2-layer MLP:  out = act2( relu( in*scale @ W1 + b1 ) @ W2 + b2 )
//   - scale = 1/max(cnt[row],1) if cnt != nullptr (fused segment-mean divide)
//   - act2 = relu if relu2 else identity
//   - W1: [64x64] row-major, W2: [64xN2] row-major, out row stride = N2
// One wave computes 16 rows x all columns with V_WMMA_F32_16X16X4_F32.
// ---------------------------------------------------------------------------
__global__ __launch_bounds__(256) void mlp2_wmma_kernel(
    const float* __restrict__ in, const float* __restrict__ cnt,
    const float* __restrict__ W1, const float* __restrict__ b1,
    const float* __restrict__ W2, const float* __restrict__ b2,
    float* __restrict__ out, int M, int N2, int relu2)
{
    __shared__ float lds[8][16 * LDSTRIDE];
    const int wave = threadIdx.x >> 5;
    const int lane = threadIdx.x & 31;
    int r0 = (blockIdx.x * 8 + wave) * 16;
    const bool active = (r0 < M);
    if (!active) r0 = M - 16;          // M is a multiple of 16 in all launches
    float* my = lds[wave];             // wave-private tile: no block barrier needed

    // ---- stage 16 input rows into LDS (coalesced float4), fusing mean-divide
    for (int i = lane; i < 256; i += 32) {
        const int r  = i >> 4;              // 0..15
        const int c4 = (i & 15) << 2;       // 0,4,...,60
        const float4 v = *(const float4*)(in + (size_t)(r0 + r) * DFEAT + c4);
        float s = 1.0f;
        if (cnt) s = 1.0f / fmaxf(cnt[r0 + r], 1.0f);
        float* p = my + r * LDSTRIDE + c4;
        p[0] = v.x * s; p[1] = v.y * s; p[2] = v.z * s; p[3] = v.w * s;
    }
    // same-wave LDS store->load: DS pipe is in-order; compiler inserts s_wait_dscnt

    const int arow = lane & 15;             // A fragment: lane holds row (lane&15)
    const int koff = (lane < 16) ? 0 : 2;   // lanes 16-31 hold K+2,K+3
    const int ncol = lane & 15;             // B/C/D fragment column within tile

    // ================= layer 1: [16x64] @ [64x64] + b1, relu =================
    v2f afr[16];
#pragma unroll
    for (int k = 0; k < 16; ++k) {
        afr[k].x = my[arow * LDSTRIDE + 4 * k + koff];
        afr[k].y = my[arow * LDSTRIDE + 4 * k + koff + 1];
    }
    v8f res1[4];
#pragma unroll
    for (int nt = 0; nt < 4; ++nt) {
        const int n0 = nt * 16;
        const float bv = b1[n0 + ncol];
        v8f acc = {bv, bv, bv, bv, bv, bv, bv, bv};
#pragma unroll
        for (int k = 0; k < 16; ++k) {
            v2f bfr;
            bfr.x = W1[(4 * k + koff) * DFEAT + n0 + ncol];
            bfr.y = W1[(4 * k + koff + 1) * DFEAT + n0 + ncol];
            acc = __builtin_amdgcn_wmma_f32_16x16x4_f32(
                false, afr[k], false, bfr, (short)0, acc, false, false);
        }
#pragma unroll
        for (int i = 0; i < 8; ++i) acc[i] = fmaxf(acc[i], 0.0f);
        res1[nt] = acc;
    }
    // write layer-1 activations back to the (now free) LDS tile, row-major
#pragma unroll
    for (int nt = 0; nt < 4; ++nt) {
#pragma unroll
        for (int i = 0; i < 8; ++i) {
            const int rr = i + ((lane < 16) ? 0 : 8);
            my[rr * LDSTRIDE + nt * 16 + ncol] = res1[nt][i];
        }
    }

    // ================= layer 2: [16x64] @ [64xN2] + b2, (relu) ===============
#pragma unroll
    for (int k = 0; k < 16; ++k) {
        afr[k].x = my[arow * LDSTRIDE + 4 * k + koff];
        afr[k].y = my[arow * LDSTRIDE + 4 * k + koff + 1];
    }
    const int ntiles2 = (N2 + 15) >> 4;
    for (int nt = 0; nt < ntiles2; ++nt) {
        const int n0 = nt * 16;
        const int n  = n0 + ncol;
        const int nc = (n < N2) ? n : (N2 - 1);   // clamp for N2=40 ragged tile
        const float bv = b2[nc];
        v8f acc = {bv, bv, bv, bv, bv, bv, bv, bv};
#pragma unroll
        for (int k = 0; k < 16; ++k) {
            v2f bfr;
            bfr.x = W2[(4 * k + koff) * N2 + nc];
            bfr.y = W2[(4 * k + koff + 1) * N2 + nc];
            acc = __builtin_amdgcn_wmma_f32_16x16x4_f32(
                false, afr[k], false, bfr, (short)0, acc, false, false);
        }
        if (active && n < N2) {
#pragma unroll
            for (int i = 0; i < 8; ++i) {
                const int rr = i + ((lane < 16) ? 0 : 8);
                float v = acc[i];
                if (relu2) v = fmaxf(v, 0.0f);
                out[(size_t)(r0 + rr) * N2 + n] = v;
            }
        }
    }
}

// ---------------------------------------------------------------------------
// Sparse propagate: agg[dst] += h[src]*norm ; cnt[dst] += 1   (16 thr / edge)
// ---------------------------------------------------------------------------
__global__ __launch_bounds__(256) void scatter_kernel(
    const float* __restrict__ h, const float* __restrict__ norm,
    const int* __restrict__ src, const int* __restrict__ dst,
    float* __restrict__ agg, float* __restrict__ cnt, int nnz)
{
    const long long t = (long long)blockIdx.x * blockDim.x + threadIdx.x;
    const long long e = t >> 4;
    if (e >= nnz) return;
    const int c = (int)(t & 15) << 2;
    const int s = src[e];
    const int d = dst[e];
    const float w = norm[e];
    const float4 v = *(const float4*)(h + (size_t)s * DFEAT + c);
    float* p = agg + (size_t)d * DFEAT + c;
    atomicAdd(p + 0, v.x * w);
    atomicAdd(p + 1, v.y * w);
    atomicAdd(p + 2, v.z * w);
    atomicAdd(p + 3, v.w * w);
    if (c == 0) atomicAdd(cnt + d, 1.0f);
}

__global__ __launch_bounds__(256) void zero_kernel(float4* __restrict__ p, int n4)
{
    const int i = blockIdx.x * blockDim.x + threadIdx.x;
    if (i < n4) p[i] = make_float4(0.f, 0.f, 0.f, 0.f);
}

// ---------------------------------------------------------------------------
extern "C" void kernel_launch(void* const* d_in, const int* in_sizes, int n_in,
                              void* d_out, int out_size, void* d_ws, size_t ws_size,
                              hipStream_t stream)
{
    (void)n_in; (void)out_size; (void)ws_size;
    const float* x    = (const float*)d_in[0];
    const float* Wc   = (const float*)d_in[1];   // [2,2,2,64,64]
    const float* bc   = (const float*)d_in[2];   // [2,2,2,64]
    const float* Wcl1 = (const float*)d_in[3];   // [64,64]
    const float* bcl1 = (const float*)d_in[4];   // [64]
    const float* Wcl2 = (const float*)d_in[5];   // [64,40]
    const float* bcl2 = (const float*)d_in[6];   // [40]
    const float* norm = (const float*)d_in[7];   // [nnz]
    const int*  V_idx = (const int*)d_in[8];     // [nnz]
    const int*  E_idx = (const int*)d_in[9];     // [nnz]

    const int NN  = 100000;   // num_nodes (fixed by reference)
    const int NE  = 50000;    // num_hyperedges
    const int NC  = 40;       // num classes
    const int nnz = in_sizes[7];

    // workspace layout (floats)
    float* ws   = (float*)d_ws;
    float* henc = ws;                  // 6,400,000  encoder outputs
    float* agg  = ws + 6400000;        // 6,400,000  scatter accumulator
    float* cnt  = ws + 12800000;       //   100,000  segment counts (contig w/ agg)
    float* h    = ws + 12900000;       // 6,400,000  conv outputs

    const float* WcP[2][2][2]; const float* bcP[2][2][2];
    for (int cv = 0; cv < 2; ++cv)
        for (int pt = 0; pt < 2; ++pt)
            for (int ly = 0; ly < 2; ++ly) {
                WcP[cv][pt][ly] = Wc + (((cv * 2) + pt) * 2 + ly) * 4096;
                bcP[cv][pt][ly] = bc + (((cv * 2) + pt) * 2 + ly) * 64;
            }

    const dim3 blk(256);
    const dim3 gNN((NN + 127) / 128), gNE((NE + 127) / 128);
    const int  n4z = (6400000 + 100000) / 4;            // agg+cnt contiguous
    const dim3 gz((n4z + 255) / 256);
    const dim3 gs((unsigned)(((long long)nnz * 16 + 255) / 256));

    // ---- conv 1: V -> E ----
    mlp2_wmma_kernel<<<gNN, blk, 0, stream>>>(x, nullptr,
        WcP[0][0][0], bcP[0][0][0], WcP[0][0][1], bcP[0][0][1], henc, NN, 64, 1);
    zero_kernel<<<gz, blk, 0, stream>>>((float4*)agg, n4z);
    scatter_kernel<<<gs, blk, 0, stream>>>(henc, norm, V_idx, E_idx, agg, cnt, nnz);
    mlp2_wmma_kernel<<<gNE, blk, 0, stream>>>(agg, cnt,
        WcP[0][1][0], bcP[0][1][0], WcP[0][1][1], bcP[0][1][1], h, NE, 64, 1);

    // ---- conv 2: E -> V ----
    mlp2_wmma_kernel<<<gNE, blk, 0, stream>>>(h, nullptr,
        WcP[1][0][0], bcP[1][0][0], WcP[1][0][1], bcP[1][0][1], henc, NE, 64, 1);
    zero_kernel<<<gz, blk, 0, stream>>>((float4*)agg, n4z);
    scatter_kernel<<<gs, blk, 0, stream>>>(henc, norm, E_idx, V_idx, agg, cnt, nnz);
    mlp2_wmma_kernel<<<gNN, blk, 0, stream>>>(agg, cnt,
        WcP[1][1][0], bcP[1][1][0], WcP[1][1][1], bcP[1][1][1], h, NN, 64, 1);

    // ---- classifier (no final relu) ----
    mlp2_wmma_kernel<<<gNN, blk, 0, stream>>>(h, nullptr,
        Wcl1, bcl1, Wcl2, bcl2, (float*)d_out, NN, NC, 0);
}